// ToFrame_57878979281422
// MI455X (gfx1250) — compile-verified
//
#include <hip/hip_runtime.h>

#define TILE 256

// ---------------- CDNA5 async global->LDS staging helpers ----------------
#if __has_builtin(__builtin_amdgcn_global_load_async_to_lds_b32)
#define HAVE_ASYNC_LDS 1
typedef __attribute__((address_space(1))) int gint_t;   // AS1: "__device__ int*"
typedef __attribute__((address_space(3))) int lint_t;   // AS3: "__shared__ int*"
#else
#define HAVE_ASYNC_LDS 0
#endif

#if __has_builtin(__builtin_amdgcn_global_load_async_to_lds_b128)
#define HAVE_ASYNC_B128 1
typedef int v4i_ __attribute__((ext_vector_type(4)));
typedef __attribute__((address_space(1))) v4i_ gv4i_t;
typedef __attribute__((address_space(3))) v4i_ lv4i_t;
#else
#define HAVE_ASYNC_B128 0
#endif

__device__ __forceinline__ void stage_b32(const float* g, float* l) {
#if HAVE_ASYNC_LDS
    __builtin_amdgcn_global_load_async_to_lds_b32((gint_t*)g, (lint_t*)l, 0, 0);
#else
    *l = *g;
#endif
}

__device__ __forceinline__ void stage_b128(const float* g, float* l) {
#if HAVE_ASYNC_B128
    // 16B async DMA-style copy global -> LDS, tracked by ASYNCcnt
    __builtin_amdgcn_global_load_async_to_lds_b128((gv4i_t*)g, (lv4i_t*)l, 0, 0);
#else
    stage_b32(g + 0, l + 0);
    stage_b32(g + 1, l + 1);
    stage_b32(g + 2, l + 2);
    stage_b32(g + 3, l + 3);
#endif
}

// Cooperatively stage ndw dwords from gbase (16B-aligned) into LDS.
__device__ __forceinline__ void stage_tile(const float* gbase, float* lbase,
                                           int ndw, int tid, int nthreads) {
    const int nq = ndw >> 2;
    for (int q = tid; q < nq; q += nthreads)
        stage_b128(gbase + 4 * q, lbase + 4 * q);
    const int rem = ndw & 3;
    if (tid < rem) {
        const int off = nq * 4 + tid;
        stage_b32(gbase + off, lbase + off);
    }
}

__device__ __forceinline__ void wait_stage() {
#if HAVE_ASYNC_LDS || HAVE_ASYNC_B128
#if __has_builtin(__builtin_amdgcn_s_wait_asynccnt)
    __builtin_amdgcn_s_wait_asynccnt(0);
#else
    asm volatile("s_wait_asynccnt 0" ::: "memory");
#endif
#endif
}

// ---------------- Kernel 1: zero output volume, init ws min/max ----------------
__global__ void zero_and_init(float4* out4, int n4, float* tail, int ntail, unsigned* ws) {
    int i = blockIdx.x * blockDim.x + threadIdx.x;
    if (i < n4) out4[i] = make_float4(0.f, 0.f, 0.f, 0.f);
    if (i < 256) {                     // slots for up to 256 batches
        ws[i]       = 0x7F800000u;     // +inf bits  (min slots)
        ws[256 + i] = 0u;              // +0.0 bits  (max slots; t >= 0)
    }
    if (i < ntail) tail[i] = 0.f;
}

// ---------------- Kernel 2: per-batch raw min/max of t ----------------
// Positive floats order identically to their uint bit patterns, so we reduce
// with uint min/max (ds_min_u32/ds_max_u32 in LDS, then global atomics).
// Note: segment_max(t - tmin + 1e-8) == fl(fl(rawmax - tmin) + 1e-8) exactly
// (monotone rounding), so raw per-batch min/max is sufficient.
__global__ void __launch_bounds__(TILE) minmax_kernel(
        const float* __restrict__ ev, int n, int perBlock,
        unsigned* __restrict__ wsmin, unsigned* __restrict__ wsmax) {
    __shared__ float    stage[TILE * 5];
    __shared__ unsigned smin[256], smax[256];
    const int tid = threadIdx.x;

    smin[tid] = 0x7F800000u;
    smax[tid] = 0u;
    __syncthreads();

    const int start = blockIdx.x * perBlock;   // perBlock is a multiple of TILE
    const int end   = min(n, start + perBlock);

    int      curb   = -1;
    unsigned curmin = 0x7F800000u, curmax = 0u;

    for (int base = start; base < end; base += TILE) {
        const int cnt = min(TILE, end - base);
        stage_tile(ev + (size_t)base * 5, stage, cnt * 5, tid, TILE);
        wait_stage();
        __syncthreads();

        if (tid < cnt) {
            const float t = stage[tid * 5 + 2];
            const int   b = (int)stage[tid * 5 + 4];
            const unsigned u = __float_as_uint(t);
            if (b != curb) {           // events sorted by b -> flushes are rare
                if (curb >= 0) {
                    atomicMin(&smin[curb], curmin);
                    atomicMax(&smax[curb], curmax);
                }
                curb = b; curmin = u; curmax = u;
            } else {
                curmin = (u < curmin) ? u : curmin;
                curmax = (u > curmax) ? u : curmax;
            }
        }
        __syncthreads();               // LDS tile reuse next iteration
    }
    if (curb >= 0) {
        atomicMin(&smin[curb], curmin);
        atomicMax(&smax[curb], curmax);
    }
    __syncthreads();

    if (smin[tid] != 0x7F800000u) {    // slot touched by this block
        atomicMin(&wsmin[tid], smin[tid]);
        atomicMax(&wsmax[tid], smax[tid]);
    }
}

// ---------------- Kernel 3: normalize, bin, scatter-add ----------------
__global__ void __launch_bounds__(TILE) scatter_kernel(
        const float* __restrict__ ev, int n,
        const unsigned* __restrict__ wsmin, const unsigned* __restrict__ wsmax,
        const int* __restrict__ pC, const int* __restrict__ pH,
        const int* __restrict__ pW, float* __restrict__ out) {
    __shared__ float stage[TILE * 5];
    __shared__ float bmin[256], bmax[256];
    const int tid = threadIdx.x;

    bmin[tid] = __uint_as_float(wsmin[tid]);
    bmax[tid] = __uint_as_float(wsmax[tid]);

    const int C = *pC, H = *pH, W = *pW;
    const int base = blockIdx.x * TILE;        // 256 events * 20B = 16B-aligned
    const int cnt  = min(TILE, n - base);
    stage_tile(ev + (size_t)base * 5, stage, cnt * 5, tid, TILE);
    wait_stage();
    __syncthreads();

    if (tid < cnt) {
        const float xf = stage[tid * 5 + 0];
        const float yf = stage[tid * 5 + 1];
        const float t  = stage[tid * 5 + 2];
        const float pf = stage[tid * 5 + 3];
        const int   b  = (int)stage[tid * 5 + 4];

        const float tmin  = bmin[b];
        const float tmaxs = (bmax[b] - tmin) + 1e-8f;   // == segment_max(t - tmin + 1e-8)
        const float ts    = (t - tmin) + 1e-8f;
        const float tn    = ts / tmaxs;

        const bool valid = (tn > 0.0f) && (tn <= 1.0f);

        const float Cf = (float)C;
        float bf = ceilf(tn * Cf) - 1.0f;
        bf = fminf(fmaxf(bf, 0.0f), Cf - 1.0f);
        const int bin = (int)bf;

        const int xi = (int)xf, yi = (int)yf, pi = (int)pf;
        const int WH  = W * H;
        const int idx = xi + W * yi + WH * (bin + C * (pi + 2 * b));

        if (valid) atomicAdd(out + idx, 1.0f);          // global_atomic_add_f32, L2-resident
    }
}

// ---------------- host-side launcher ----------------
extern "C" void kernel_launch(void* const* d_in, const int* in_sizes, int n_in,
                              void* d_out, int out_size, void* d_ws, size_t ws_size,
                              hipStream_t stream) {
    const float* ev = (const float*)d_in[0];
    // d_in[1] = B (unused; batch ids are read from the event records)
    const int* pC = (const int*)d_in[2];
    const int* pH = (const int*)d_in[3];
    const int* pW = (const int*)d_in[4];

    const int n = in_sizes[0] / 5;
    unsigned* ws = (unsigned*)d_ws;       // [0,256)=min bits, [256,512)=max bits

    const int n4    = out_size / 4;
    const int ntail = out_size - n4 * 4;
    int zelems  = (n4 > 256) ? n4 : 256;
    int zblocks = (zelems + 255) / 256;
    zero_and_init<<<zblocks, 256, 0, stream>>>((float4*)d_out, n4,
                                               (float*)d_out + (size_t)n4 * 4, ntail, ws);

    // perBlock rounded up to a multiple of TILE so every tile base is
    // 256-event-aligned => byte offset multiple of 5120 => 16B aligned for b128.
    const int mblocks   = 1024;
    int perBlock = (n + mblocks * TILE - 1) / (mblocks * TILE);
    perBlock *= TILE;
    minmax_kernel<<<mblocks, TILE, 0, stream>>>(ev, n, perBlock, ws, ws + 256);

    const int sblocks = (n + TILE - 1) / TILE;
    scatter_kernel<<<sblocks, TILE, 0, stream>>>(ev, n, ws, ws + 256, pC, pH, pW,
                                                 (float*)d_out);
}